// SurgeMModel_54898271977821
// MI455X (gfx1250) — compile-verified
//
#include <hip/hip_runtime.h>
#include <hip/hip_bf16.h>
#include <cstdint>

// Problem constants (fixed by reference setup_inputs)
#define BB 4
#define TT 4096
#define DD 1024
#define CC 64
#define CSZ 64

typedef float v2f __attribute__((ext_vector_type(2)));
typedef float v8f __attribute__((ext_vector_type(8)));
typedef int   v4i __attribute__((ext_vector_type(4)));

#define AS1 __attribute__((address_space(1)))
#define AS3 __attribute__((address_space(3)))

// D = A(16x4) x B(4x16) + C, fp32. Keep all modifier args as literals.
#define WMMA_F32X4(A_, B_, C_) \
  __builtin_amdgcn_wmma_f32_16x16x4_f32(false, (A_), false, (B_), (short)0, (C_), false, false)

// ---------------------------------------------------------------------------
// Async global -> LDS copy of 16 bytes (GLOBAL_LOAD_ASYNC_TO_LDS_B128,
// ASYNCcnt-tracked).  Builtin when available, inline asm otherwise.
// Builtin takes pointers to 16-byte int vectors (per hipcc diagnostic).
// ---------------------------------------------------------------------------
__device__ __forceinline__ void async_cp16(const float* g, float* l) {
#if __has_builtin(__builtin_amdgcn_global_load_async_to_lds_b128)
  __builtin_amdgcn_global_load_async_to_lds_b128(
      (AS1 v4i*)(uintptr_t)g, (AS3 v4i*)l, /*offset=*/0, /*cpol=*/0);
#else
  const unsigned lds = (unsigned)(uintptr_t)(AS3 void*)l;
  const unsigned long long ga = (unsigned long long)(uintptr_t)g;
  asm volatile("global_load_async_to_lds_b128 %0, %1, off"
               :: "v"(lds), "v"(ga) : "memory");
#endif
}

__device__ __forceinline__ void wait_async0() {
#if __has_builtin(__builtin_amdgcn_s_wait_asynccnt)
  __builtin_amdgcn_s_wait_asynccnt(0);
#else
  asm volatile("s_wait_asynccnt 0x0" ::: "memory");
#endif
}

// ---------------------------------------------------------------------------
// Kernel 1: Z = X @ W0^T   (rows = B*T = 16384, cols = D = 1024, K = D)
// 64x64 tile per workgroup, 8 waves, 2 subtiles/wave.
// Double-buffered async LDS staging (32-wide K stages).
// ---------------------------------------------------------------------------
__global__ __launch_bounds__(256) void k_zgemm(const float* __restrict__ X,
                                               const float* __restrict__ W0,
                                               float* __restrict__ Z) {
  __shared__ float sA[2][64][36];   // 36-float pitch: conflict-free, 16B aligned
  __shared__ float sB[2][64][36];
  const int r0  = blockIdx.x * 64;   // row tile over B*T
  const int n0  = blockIdx.y * 64;   // col tile over D
  const int tid = threadIdx.x;
  const int w   = tid >> 5;
  const int l   = tid & 31;
  const int lm  = l & 15;            // row/col within subtile
  const int lh  = l >> 4;            // k-half select (K0/K1 vs K2/K3)
  const int sr  = (w & 3) * 16;      // subtile row
  const int scB = (w >> 2) * 2;      // subtile col pair base (0 or 2)

  v8f acc0 = {};
  v8f acc1 = {};

  // issue one 64x32 stage of both operands into buffer `buf`
  auto issue_stage = [&](int buf, int kk) {
#pragma unroll
    for (int rep = 0; rep < 2; ++rep) {
      const int f   = tid + rep * 256;
      const int row = f >> 3;
      const int kq  = (f & 7) << 2;
      async_cp16(X + (size_t)(r0 + row) * DD + kk + kq, &sA[buf][row][kq]);
      async_cp16(W0 + (size_t)(n0 + row) * DD + kk + kq, &sB[buf][row][kq]);
    }
  };

  issue_stage(0, 0);
  for (int kk = 0, it = 0; kk < DD; kk += 32, ++it) {
    const int cur = it & 1;
    const int nxt = cur ^ 1;
    wait_async0();        // my stage-`cur` async loads have landed in LDS
    __syncthreads();      // everyone's have; also: all reads of buf `nxt` done
    if (kk + 32 < DD) issue_stage(nxt, kk + 32);

#pragma unroll
    for (int k4 = 0; k4 < 32; k4 += 4) {
      v2f a, b0, b1;
      a.x  = sA[cur][sr + lm][k4 + lh * 2 + 0];
      a.y  = sA[cur][sr + lm][k4 + lh * 2 + 1];
      b0.x = sB[cur][scB * 16 + lm][k4 + lh * 2 + 0];
      b0.y = sB[cur][scB * 16 + lm][k4 + lh * 2 + 1];
      b1.x = sB[cur][(scB + 1) * 16 + lm][k4 + lh * 2 + 0];
      b1.y = sB[cur][(scB + 1) * 16 + lm][k4 + lh * 2 + 1];
      acc0 = WMMA_F32X4(a, b0, acc0);
      acc1 = WMMA_F32X4(a, b1, acc1);
    }
  }

  // Writeback: vgpr j of lane l holds (row = j + 8*lh, col = lm)
#pragma unroll
  for (int j = 0; j < 8; ++j) {
    const int m = sr + j + 8 * lh;
    Z[(size_t)(r0 + m) * DD + n0 + scB * 16 + lm]       = acc0[j];
    Z[(size_t)(r0 + m) * DD + n0 + (scB + 1) * 16 + lm] = acc1[j];
  }
}

// ---------------------------------------------------------------------------
// Kernel 2: decay/scale table  S[b][c][i] = g_i * (prod_{k=i+1}^{c-1} a_k)^2
// (zero for i >= c).  One thread per (b, c).
// ---------------------------------------------------------------------------
__global__ void k_scales(const float* __restrict__ g,
                         const float* __restrict__ a,
                         float* __restrict__ S) {
  const int t = threadIdx.x;         // 0..255
  const int b = t >> 6;
  const int c = t & 63;
  float* Srow = S + ((size_t)(b * CC + c)) * CC;
  for (int i = 0; i < CC; ++i) Srow[i] = 0.0f;
  float p = 1.0f;                    // running product of a's, squared below
  for (int i = c - 1; i >= 0; --i) {
    Srow[i] = g[b * CC + i] * p * p;
    p *= a[b * CC + i];
  }
}

// ---------------------------------------------------------------------------
// Kernel 3: Gram matrix G = U @ V^T per batch (WMMA), then forward
// substitution  m_c = u_c + sum_{i<c} S[c][i]*G[c][i] * m_i.
// One workgroup (8 waves) per batch.
// ---------------------------------------------------------------------------
__global__ __launch_bounds__(256) void k_m(const float* __restrict__ U,
                                           const float* __restrict__ V,
                                           const float* __restrict__ S,
                                           float* __restrict__ M) {
  __shared__ float sG[64][65];
  __shared__ float coef[64];
  const int b   = blockIdx.x;
  const int tid = threadIdx.x;
  const int w   = tid >> 5;
  const int l   = tid & 31;
  const int lm  = l & 15;
  const int lh  = l >> 4;
  const int sr  = (w & 3) * 16;
  const int scB = (w >> 2) * 2;

  const float* Ub = U + (size_t)b * CC * DD;
  const float* Vb = V + (size_t)b * CC * DD;

  v8f acc0 = {};
  v8f acc1 = {};
  for (int k = 0; k < DD; k += 4) {
    v2f a, b0, b1;
    a.x  = Ub[(size_t)(sr + lm) * DD + k + lh * 2 + 0];
    a.y  = Ub[(size_t)(sr + lm) * DD + k + lh * 2 + 1];
    b0.x = Vb[(size_t)(scB * 16 + lm) * DD + k + lh * 2 + 0];
    b0.y = Vb[(size_t)(scB * 16 + lm) * DD + k + lh * 2 + 1];
    b1.x = Vb[(size_t)((scB + 1) * 16 + lm) * DD + k + lh * 2 + 0];
    b1.y = Vb[(size_t)((scB + 1) * 16 + lm) * DD + k + lh * 2 + 1];
    acc0 = WMMA_F32X4(a, b0, acc0);
    acc1 = WMMA_F32X4(a, b1, acc1);
  }
#pragma unroll
  for (int j = 0; j < 8; ++j) {
    const int m = sr + j + 8 * lh;
    sG[m][scB * 16 + lm]       = acc0[j];
    sG[m][(scB + 1) * 16 + lm] = acc1[j];
  }
  __syncthreads();

  // Sequential forward substitution over chunks; M kept in global scratch
  // (L2-resident).  volatile to bypass stale L0 lines across waves.
  volatile float* Mb = M + (size_t)b * CC * DD;
  const float* Sb = S + (size_t)b * CC * CC;
  for (int c = 0; c < CC; ++c) {
    if (tid < 64) coef[tid] = (tid < c) ? Sb[c * CC + tid] * sG[c][tid] : 0.0f;
    __syncthreads();
#pragma unroll
    for (int rep = 0; rep < 4; ++rep) {
      const int dd = tid + rep * 256;
      float acc = Ub[(size_t)c * DD + dd];
      for (int i = 0; i < c; ++i) acc += coef[i] * Mb[(size_t)i * DD + dd];
      Mb[(size_t)c * DD + dd] = acc;
    }
    __threadfence();
    __syncthreads();
  }
}

// ---------------------------------------------------------------------------
// Kernel 4: per (b, c):  T1 = (Z_c @ V^T) * S[c][.],  corr = T1 @ M.
// One workgroup per (b, c); both GEMMs via WMMA; async-staged GEMM-1.
// ---------------------------------------------------------------------------
__global__ __launch_bounds__(256) void k_corr(const float* __restrict__ Z,
                                              const float* __restrict__ V,
                                              const float* __restrict__ S,
                                              const float* __restrict__ M,
                                              float* __restrict__ Out) {
  __shared__ float sA[2][64][36];
  __shared__ float sB[2][64][36];
  __shared__ float sT[64][66];   // scaled T1, even pitch for b64 reads
  const int bc  = blockIdx.x;    // b*CC + c
  const int b   = bc >> 6;
  const int c   = bc & 63;
  const int tid = threadIdx.x;
  const int w   = tid >> 5;
  const int l   = tid & 31;
  const int lm  = l & 15;
  const int lh  = l >> 4;
  const int sr  = (w & 3) * 16;
  const int scB = (w >> 2) * 2;

  const float* Zc = Z + ((size_t)(b * TT + c * CSZ)) * DD;  // 64 x D
  const float* Vb = V + (size_t)b * CC * DD;
  const float* Sb = S + (size_t)bc * CC;
  const float* Mb = M + (size_t)b * CC * DD;

  auto issue_stage = [&](int buf, int kk) {
#pragma unroll
    for (int rep = 0; rep < 2; ++rep) {
      const int f   = tid + rep * 256;
      const int row = f >> 3;
      const int kq  = (f & 7) << 2;
      async_cp16(Zc + (size_t)row * DD + kk + kq, &sA[buf][row][kq]);
      async_cp16(Vb + (size_t)row * DD + kk + kq, &sB[buf][row][kq]);
    }
  };

  // ---- GEMM 1: T1 = Z_c @ V^T (64 x 64, K = D) ----
  v8f acc0 = {};
  v8f acc1 = {};
  issue_stage(0, 0);
  for (int kk = 0, it = 0; kk < DD; kk += 32, ++it) {
    const int cur = it & 1;
    const int nxt = cur ^ 1;
    wait_async0();
    __syncthreads();
    if (kk + 32 < DD) issue_stage(nxt, kk + 32);

#pragma unroll
    for (int k4 = 0; k4 < 32; k4 += 4) {
      v2f a, b0, b1;
      a.x  = sA[cur][sr + lm][k4 + lh * 2 + 0];
      a.y  = sA[cur][sr + lm][k4 + lh * 2 + 1];
      b0.x = sB[cur][scB * 16 + lm][k4 + lh * 2 + 0];
      b0.y = sB[cur][scB * 16 + lm][k4 + lh * 2 + 1];
      b1.x = sB[cur][(scB + 1) * 16 + lm][k4 + lh * 2 + 0];
      b1.y = sB[cur][(scB + 1) * 16 + lm][k4 + lh * 2 + 1];
      acc0 = WMMA_F32X4(a, b0, acc0);
      acc1 = WMMA_F32X4(a, b1, acc1);
    }
  }

  // Scale columns by S[c][i] (zero for i >= c => causal mask for free)
  __syncthreads();
  const float s0 = Sb[scB * 16 + lm];
  const float s1 = Sb[(scB + 1) * 16 + lm];
#pragma unroll
  for (int j = 0; j < 8; ++j) {
    const int m = sr + j + 8 * lh;
    sT[m][scB * 16 + lm]       = acc0[j] * s0;
    sT[m][(scB + 1) * 16 + lm] = acc1[j] * s1;
  }
  __syncthreads();

  // ---- GEMM 2: corr = T1(64x64) @ M(64xD); 32 output subtiles per wave ----
  const int colGroup = (w >> 2);  // 0/1 -> halves of the 1024 output cols
  float* Oc = Out + ((size_t)(b * TT + c * CSZ)) * DD;
  for (int jt = 0; jt < 32; ++jt) {
    const int n0 = (colGroup * 32 + jt) * 16;
    v8f acc = {};
#pragma unroll
    for (int k = 0; k < 64; k += 4) {
      v2f a, bb;
      a.x  = sT[sr + lm][k + lh * 2 + 0];
      a.y  = sT[sr + lm][k + lh * 2 + 1];
      bb.x = Mb[(size_t)(k + lh * 2 + 0) * DD + n0 + lm];
      bb.y = Mb[(size_t)(k + lh * 2 + 1) * DD + n0 + lm];
      acc = WMMA_F32X4(a, bb, acc);
    }
#pragma unroll
    for (int j = 0; j < 8; ++j) {
      const int m = sr + j + 8 * lh;
      Oc[(size_t)m * DD + n0 + lm] = acc[j];
    }
  }
}

// ---------------------------------------------------------------------------
extern "C" void kernel_launch(void* const* d_in, const int* in_sizes, int n_in,
                              void* d_out, int out_size, void* d_ws, size_t ws_size,
                              hipStream_t stream) {
  (void)in_sizes; (void)n_in; (void)out_size; (void)ws_size;
  const float* X  = (const float*)d_in[0];  // attn_pre_wo (B,T,d)
  const float* U  = (const float*)d_in[1];  // u (B,C,d)
  const float* Vv = (const float*)d_in[2];  // v (B,C,d)
  const float* G  = (const float*)d_in[3];  // g (B,C,1)
  const float* A  = (const float*)d_in[4];  // a (B,C,1)
  const float* W0 = (const float*)d_in[5];  // W_0 (d,d)
  float* Out = (float*)d_out;

  // Workspace layout (floats): Z (B*T*D) | S (B*C*C) | M (B*C*D)
  float* Z = (float*)d_ws;
  float* S = Z + (size_t)BB * TT * DD;
  float* M = S + (size_t)BB * CC * CC;

  k_zgemm<<<dim3((BB * TT) / 64, DD / 64), 256, 0, stream>>>(X, W0, Z);
  k_scales<<<1, BB * CC, 0, stream>>>(G, A, S);
  k_m<<<BB, 256, 0, stream>>>(U, Vv, S, M);
  k_corr<<<BB * CC, 256, 0, stream>>>(Z, Vv, S, M, Out);
}